// Layer_FISTA_73632919323408
// MI455X (gfx1250) — compile-verified
//
#include <hip/hip_runtime.h>

typedef __attribute__((ext_vector_type(2))) float v2f;
typedef __attribute__((ext_vector_type(8))) float v8f;

#define GAMMA0 0.1f

constexpr int Bn = 16, Cn = 4, NP = 256, XD = 128, UD = 32, T = 20, PD = 64;
constexpr int NCOL = NP * T;          // 5120 columns per (b,c) GEMM
constexpr int NT2  = NCOL / 32;       // 160 double-N-tiles (32 columns each)

// Column n -> pointer/stride into prev[b,c] (time-shifted X with x0 at t=0)
__device__ __forceinline__ void prev_col(const float* __restrict__ X,
                                         const float* __restrict__ x0,
                                         int bc, int n,
                                         const float*& ptr, int& js,
                                         int& p, int& t)
{
    p = n / T;
    t = n - p * T;
    if (t == 0) {                               // prev(t=0) = x0
        ptr = x0 + ((size_t)(bc * NP + p)) * XD;
        js  = 1;
    } else {                                    // prev(t) = X(t-1)
        ptr = X + (((size_t)(bc * NP + p)) * XD) * T + (t - 1);
        js  = T;
    }
}

// ---------------------------------------------------------------------------
// Kernel 1: X_pred[b,c,p,i,t] = sum_j A[c,i,j] * prev[b,c,p,j,t]
// One wave -> full M=128 strip x TWO 16-wide N tiles (32 columns).
// Per k-step: 8 A-frag b64 loads feed 16 WMMAs (A reuse across both N tiles).
// ---------------------------------------------------------------------------
__global__ void __launch_bounds__(256)
xpred_kernel(const float* __restrict__ X, const float* __restrict__ x0,
             const float* __restrict__ A, float* __restrict__ Xpred)
{
    const int lane  = threadIdx.x & 31;
    const int wave  = threadIdx.x >> 5;
    const int strip = blockIdx.x * 8 + wave;    // 64*160 strips total

    const int nt2 = strip % NT2;
    const int bc  = strip / NT2;                // b*4 + c
    const int c   = bc & 3;

    const int lhalf = lane >> 4;                // 0: K%4 in {0,1}, 1: {2,3}
    const int l15   = lane & 15;

    // A fragment base: row l15; M-tile mt is mt*16*XD further (imm offset).
    const float* Aptr = A + ((size_t)(c * XD + l15)) * XD + (lhalf << 1);

    // Two B columns: n0 and n0+16
    const int n0 = (nt2 << 5) + l15;
    const float *Bp0, *Bp1;
    int js0, js1, p0, t0, p1, t1;
    prev_col(X, x0, bc, n0,      Bp0, js0, p0, t0);
    prev_col(X, x0, bc, n0 + 16, Bp1, js1, p1, t1);
    Bp0 += (size_t)(lhalf << 1) * js0;
    Bp1 += (size_t)(lhalf << 1) * js1;

    v8f acc0[8] = {};
    v8f acc1[8] = {};
    #pragma unroll 2
    for (int k0 = 0; k0 < XD; k0 += 4) {
        v2f b0, b1;
        b0.x = Bp0[0];   b0.y = Bp0[js0];
        b1.x = Bp1[0];   b1.y = Bp1[js1];
        #pragma unroll
        for (int mt = 0; mt < 8; ++mt) {
            v2f a;
            a.x = Aptr[mt * 16 * XD];           // b64 load, immediate offset
            a.y = Aptr[mt * 16 * XD + 1];
            acc0[mt] = __builtin_amdgcn_wmma_f32_16x16x4_f32(
                           false, a, false, b0, (short)0, acc0[mt], false, false);
            acc1[mt] = __builtin_amdgcn_wmma_f32_16x16x4_f32(
                           false, a, false, b1, (short)0, acc1[mt], false, false);
        }
        Aptr += 4;
        Bp0  += (size_t)4 * js0;
        Bp1  += (size_t)4 * js1;
    }

    // Store: acc[mt][r] -> row i = mt*16 + r + lhalf*8, fixed column (p,t)
    float* o0 = Xpred + (((size_t)(bc * NP + p0)) * XD + (lhalf << 3)) * T + t0;
    float* o1 = Xpred + (((size_t)(bc * NP + p1)) * XD + (lhalf << 3)) * T + t1;
    #pragma unroll
    for (int mt = 0; mt < 8; ++mt)
        #pragma unroll
        for (int r = 0; r < 8; ++r) {
            o0[(size_t)(mt * 16 + r) * T] = acc0[mt][r];
            o1[(size_t)(mt * 16 + r) * T] = acc1[mt][r];
        }
}

// ---------------------------------------------------------------------------
// Kernel 2: recon[u] = sum_j Cm[c,u,j] * X[b,c,p,j,t]  (u in [0,64))
// then pixel-shuffle: p=gy*16+gx, u=py*8+px -> img[y=gy*8+py][x=gx*8+px]
// One wave -> full M=64 strip x two 16-wide N tiles.
// ---------------------------------------------------------------------------
__global__ void __launch_bounds__(256)
img_kernel(const float* __restrict__ X, const float* __restrict__ Cm,
           float* __restrict__ img)
{
    const int lane  = threadIdx.x & 31;
    const int wave  = threadIdx.x >> 5;
    const int strip = blockIdx.x * 8 + wave;    // 64*160 strips total

    const int nt2 = strip % NT2;
    const int bc  = strip / NT2;
    const int c   = bc & 3;

    const int lhalf = lane >> 4;
    const int l15   = lane & 15;

    const float* Aptr = Cm + ((size_t)(c * PD + l15)) * XD + (lhalf << 1);

    const int n0 = (nt2 << 5) + l15;
    const int p0 = n0 / T,        t0 = n0 - p0 * T;
    const int n1 = n0 + 16;
    const int p1 = n1 / T,        t1 = n1 - p1 * T;
    const float* Bp0 = X + (((size_t)(bc * NP + p0)) * XD) * T + t0
                         + (size_t)(lhalf << 1) * T;
    const float* Bp1 = X + (((size_t)(bc * NP + p1)) * XD) * T + t1
                         + (size_t)(lhalf << 1) * T;

    v8f acc0[4] = {};
    v8f acc1[4] = {};
    #pragma unroll 2
    for (int k0 = 0; k0 < XD; k0 += 4) {
        v2f b0, b1;
        b0.x = Bp0[0];  b0.y = Bp0[T];
        b1.x = Bp1[0];  b1.y = Bp1[T];
        #pragma unroll
        for (int mt = 0; mt < 4; ++mt) {
            v2f a;
            a.x = Aptr[mt * 16 * XD];
            a.y = Aptr[mt * 16 * XD + 1];
            acc0[mt] = __builtin_amdgcn_wmma_f32_16x16x4_f32(
                           false, a, false, b0, (short)0, acc0[mt], false, false);
            acc1[mt] = __builtin_amdgcn_wmma_f32_16x16x4_f32(
                           false, a, false, b1, (short)0, acc1[mt], false, false);
        }
        Aptr += 4;
        Bp0  += (size_t)4 * T;
        Bp1  += (size_t)4 * T;
    }

    // Scatter-store with the patch->image shuffle folded in.
    const int gy0 = p0 >> 4, gx0 = p0 & 15;
    const int gy1 = p1 >> 4, gx1 = p1 & 15;
    #pragma unroll
    for (int mt = 0; mt < 4; ++mt) {
        #pragma unroll
        for (int r = 0; r < 8; ++r) {
            const int u  = mt * 16 + r + (lhalf << 3);
            const int py = u >> 3, px = u & 7;
            img[(((size_t)(bc * 128 + (gy0 << 3) + py)) * 128
                 + (gx0 << 3) + px) * T + t0] = acc0[mt][r];
            img[(((size_t)(bc * 128 + (gy1 << 3) + py)) * 128
                 + (gx1 << 3) + px) * T + t1] = acc1[mt][r];
        }
    }
}

// ---------------------------------------------------------------------------
// Kernel 3: Bu[i,t] = sum_j Bm[c,i,j]*U[b,c,j,t];
//           X_U = GAMMA0*(1+exp(-Bu))/2 * sum_p |X[b,c,p,i,t]|
// ---------------------------------------------------------------------------
__global__ void __launch_bounds__(256)
xu_kernel(const float* __restrict__ X, const float* __restrict__ U,
          const float* __restrict__ Bm, float* __restrict__ XU)
{
    const int idx = blockIdx.x * 256 + threadIdx.x;   // 16*4*128*20 = 163840
    const int t   = idx % T;
    int tmp       = idx / T;
    const int i   = tmp % XD;
    const int bc  = tmp / XD;
    const int c   = bc & 3;

    const float* Brow = Bm + (size_t)(c * XD + i) * UD;
    const float* Ucol = U + (size_t)bc * UD * T + t;
    float bu = 0.f;
    #pragma unroll
    for (int j = 0; j < UD; ++j)
        bu += Brow[j] * Ucol[(size_t)j * T];
    const float gamma = GAMMA0 * (1.0f + __expf(-bu)) * 0.5f;

    const float* Xp = X + ((size_t)bc * NP * XD + i) * T + t;
    float s = 0.f;
    for (int p = 0; p < NP; ++p)
        s += fabsf(Xp[(size_t)p * XD * T]);

    XU[idx] = gamma * s;
}

// ---------------------------------------------------------------------------
extern "C" void kernel_launch(void* const* d_in, const int* in_sizes, int n_in,
                              void* d_out, int out_size, void* d_ws, size_t ws_size,
                              hipStream_t stream)
{
    const float* X  = (const float*)d_in[0];
    const float* U  = (const float*)d_in[1];
    const float* x0 = (const float*)d_in[2];
    const float* A  = (const float*)d_in[3];
    const float* Bm = (const float*)d_in[4];
    const float* Cm = (const float*)d_in[5];

    float* out   = (float*)d_out;
    float* img   = out;                                           // 16*4*128*128*20
    float* Xpred = img + (size_t)Bn * Cn * 128 * 128 * T;         // 16*4*256*128*20
    float* XU    = Xpred + (size_t)Bn * Cn * NP * XD * T;         // 16*4*128*20

    // X_pred: 64 (bc) * 160 (double N-tiles) waves, 8 waves/block
    xpred_kernel<<<(Bn * Cn * NT2) / 8, 256, 0, stream>>>(X, x0, A, Xpred);

    // img: 64 * 160 waves
    img_kernel<<<(Bn * Cn * NT2) / 8, 256, 0, stream>>>(X, Cm, img);

    // X_U: 163840 threads
    xu_kernel<<<(Bn * Cn * XD * T) / 256, 256, 0, stream>>>(X, U, Bm, XU);
}